// SPFlowNet_82446192214594
// MI455X (gfx1250) — compile-verified
//
#include <hip/hip_runtime.h>
#include <math.h>

// ---------------------------------------------------------------------------
// SPFlowNet forward for MI455X (gfx1250, wave32, WMMA + TDM).
// fp32 end-to-end; matrix math on V_WMMA_F32_16X16X4_F32; LDS staging of
// weight/GEMM tiles via TENSOR_LOAD_TO_LDS (TDM) when the builtin exists.
// ---------------------------------------------------------------------------

typedef __attribute__((ext_vector_type(2))) float v2f;
typedef __attribute__((ext_vector_type(8))) float v8f;

#define DEV_INLINE __device__ __forceinline__

constexpr int BB = 2;          // batch
constexpr int NN = 4096;       // points
constexpr int KN = 16;         // graph knn == corr knn
constexpr int CH = 128;        // feature width
constexpr long long BN = (long long)BB * NN;

DEV_INLINE v8f wmma4(v2f a, v2f b, v8f c) {
  // D(16x16,f32) = A(16x4,f32) * B(4x16,f32) + C
  return __builtin_amdgcn_wmma_f32_16x16x4_f32(
      /*neg_a=*/false, a, /*neg_b=*/false, b,
      /*c_mod=*/(short)0, c, /*reuse_a=*/false, /*reuse_b=*/false);
}

// ---------------------------------------------------------------------------
// Tensor Data Mover: one-descriptor 1-D global->LDS copy (fp32 elements).
// Issued by a single wave per block; completion via s_wait_tensorcnt.
// ---------------------------------------------------------------------------
#if __has_builtin(__builtin_amdgcn_tensor_load_to_lds)
#define HAVE_TDM 1
typedef __attribute__((ext_vector_type(4))) unsigned int u32x4;
typedef __attribute__((ext_vector_type(4))) int i32x4;
typedef __attribute__((ext_vector_type(8))) int i32x8;

DEV_INLINE void tdm_load_1d(void* ldsDst, const void* gsrc, unsigned nElem) {
  const unsigned lds_off = (unsigned)(unsigned long long)ldsDst; // addr[31:0] = LDS byte offset
  const unsigned long long ga = (unsigned long long)gsrc;
  u32x4 g0;
  g0[0] = 1u;                                            // count=1 (valid user D#)
  g0[1] = lds_off;                                       // lds_addr
  g0[2] = (unsigned)(ga & 0xFFFFFFFFull);                // global_addr[31:0]
  g0[3] = (unsigned)((ga >> 32) & 0x01FFFFFFull)         // global_addr[56:32]
          | (2u << 30);                                  // type = 2 ("image")
  i32x8 g1;
  g1[0] = (int)(2u << 16);                               // data_size = 4 bytes
  g1[1] = (int)((nElem & 0xFFFFu) << 16);                // tensor_dim0[15:0]
  g1[2] = (int)((nElem >> 16) & 0xFFFFu) | (1 << 16);    // tensor_dim0[31:16] | tensor_dim1=1
  g1[3] = (int)(nElem << 16);                            // tile_dim0 = nElem (<= 65535)
  g1[4] = 1;                                             // tile_dim1 = 1, tile_dim2 = 0
  g1[5] = (int)nElem;                                    // tensor_dim0_stride[31:0]
  g1[6] = 0;                                             // stride hi | dim1_stride lo
  g1[7] = 0;
  const i32x4 gz = {0, 0, 0, 0};
#if defined(__clang_major__) && (__clang_major__ >= 23)
  const i32x8 gz8 = {0, 0, 0, 0, 0, 0, 0, 0};
  __builtin_amdgcn_tensor_load_to_lds(g0, g1, gz, gz, gz8, 0);
#else
  __builtin_amdgcn_tensor_load_to_lds(g0, g1, gz, gz, 0);
#endif
}
DEV_INLINE void tdm_wait() { __builtin_amdgcn_s_wait_tensorcnt(0); }
#else
#define HAVE_TDM 0
#endif

// ---------------------------------------------------------------------------
// KNN: one thread per query, register insertion sort (fully unrolled).
// ---------------------------------------------------------------------------
template<int KK>
__global__ __launch_bounds__(256) void knn_kernel(
    const float* __restrict__ qpc, const float* __restrict__ rpc,
    int* __restrict__ out)
{
  long long t = (long long)blockIdx.x * blockDim.x + threadIdx.x;
  if (t >= BN) return;
  const int b = (int)(t / NN);
  const float* q = qpc + t * 3;
  const float qx = q[0], qy = q[1], qz = q[2];
  float bd[KK]; int bi[KK];
#pragma unroll
  for (int i = 0; i < KK; ++i) { bd[i] = 3.4e38f; bi[i] = 0; }
  const float* rb = rpc + (long long)b * NN * 3;
  for (int j = 0; j < NN; ++j) {
    const float dx = rb[3*j] - qx, dy = rb[3*j+1] - qy, dz = rb[3*j+2] - qz;
    const float d = dx*dx + dy*dy + dz*dz;
    if (d < bd[KK-1]) {
      int pos = 0;
#pragma unroll
      for (int i = 0; i < KK; ++i) pos += (bd[i] <= d) ? 1 : 0;
#pragma unroll
      for (int i = KK-1; i > 0; --i) if (i > pos) { bd[i] = bd[i-1]; bi[i] = bi[i-1]; }
#pragma unroll
      for (int i = 0; i < KK; ++i) if (i == pos) { bd[i] = d; bi[i] = j; }
    }
  }
#pragma unroll
  for (int i = 0; i < KK; ++i) out[t * KK + i] = bi[i];
}

// ---------------------------------------------------------------------------
// PointWarping: 3-NN inverse-distance interpolation of -flowA onto pcB.
// ---------------------------------------------------------------------------
__global__ __launch_bounds__(256) void warp3_kernel(
    const float* __restrict__ pcB, const float* __restrict__ pcA,
    const float* __restrict__ flowA, float* __restrict__ outW)
{
  long long t = (long long)blockIdx.x * blockDim.x + threadIdx.x;
  if (t >= BN) return;
  const int b = (int)(t / NN);
  const float qx = pcB[t*3], qy = pcB[t*3+1], qz = pcB[t*3+2];
  const float* ab = pcA + (long long)b * NN * 3;
  const float* fb = flowA + (long long)b * NN * 3;
  float d0 = 3.4e38f, d1 = 3.4e38f, d2 = 3.4e38f;
  int i0 = 0, i1 = 0, i2 = 0;
  for (int j = 0; j < NN; ++j) {
    const float wx = ab[3*j]   + fb[3*j];
    const float wy = ab[3*j+1] + fb[3*j+1];
    const float wz = ab[3*j+2] + fb[3*j+2];
    const float dx = wx - qx, dy = wy - qy, dz = wz - qz;
    const float d = dx*dx + dy*dy + dz*dz;
    if (d < d2) {
      if (d < d1) {
        d2 = d1; i2 = i1;
        if (d < d0) { d1 = d0; i1 = i0; d0 = d; i0 = j; }
        else        { d1 = d;  i1 = j; }
      } else { d2 = d; i2 = j; }
    }
  }
  float w0 = 1.f / fmaxf(d0, 1e-10f);
  float w1 = 1.f / fmaxf(d1, 1e-10f);
  float w2 = 1.f / fmaxf(d2, 1e-10f);
  const float ws = w0 + w1 + w2;
  w0 /= ws; w1 /= ws; w2 /= ws;
#pragma unroll
  for (int c = 0; c < 3; ++c)
    outW[t*3 + c] = -(w0 * fb[3*i0 + c] + w1 * fb[3*i1 + c] + w2 * fb[3*i2 + c]);
}

// ---------------------------------------------------------------------------
// Edge-MLP engine (WMMA).  One wave == one query point == 16 edge rows.
//   MODE 0: rows = [gather(x0,idx), rel]        -> relu -> max over rows (SetConv)
//   MODE 1: rows = [gather(x0,idx), x1[q], rel] -> relu -> per-edge out (corr1)
//   MODE 2: rows = dense per-edge x0            -> relu -> per-edge out (corr2)
//   MODE 3: rows = dense per-edge x0            -> relu -> max over rows (corr3)
// Weights (Cin x COUT fp32, zero-padded to Cin4 rows) staged in LDS per block
// via the Tensor Data Mover (one descriptor per block).
// ---------------------------------------------------------------------------
template<int COUT, int MODE>
__global__ __launch_bounds__(256) void edge_mlp(
    const float* __restrict__ x0, const float* __restrict__ x1,
    const float* __restrict__ pcn, const float* __restrict__ pcq,
    const int* __restrict__ idx, const float* __restrict__ W,
    const float* __restrict__ bias, float* __restrict__ out,
    int Cx, int Cin)
{
  extern __shared__ float lds[];
  const int Cin4 = (Cin + 3) & ~3;
  float* Wl = lds;                          // Cin4 x COUT
  float* Bl = lds + (size_t)Cin4 * COUT;    // COUT

  const int tid = threadIdx.x;
  const int total = Cin4 * COUT;
#if HAVE_TDM
  if (tid < 32) tdm_load_1d(Wl, W, (unsigned)(Cin * COUT));   // wave 0 issues DMA
  for (int t = Cin * COUT + tid; t < total; t += 256) Wl[t] = 0.0f;  // pad rows
  for (int t = tid; t < COUT; t += 256) Bl[t] = bias[t];
  if (tid < 32) tdm_wait();
#else
  for (int t = tid; t < total; t += 256) {
    const int row = t / COUT;
    Wl[t] = (row < Cin) ? W[t] : 0.0f;
  }
  for (int t = tid; t < COUT; t += 256) Bl[t] = bias[t];
#endif
  __syncthreads();

  const int wave = tid >> 5;
  const int lane = tid & 31;
  const int lo = lane & 15;
  const int hi = lane >> 4;
  const long long qq = (long long)blockIdx.x * 8 + wave;   // 0 .. B*N-1
  const int b  = (int)(qq / NN);
  const int qi = (int)(qq % NN);

  float r0 = 0.f, r1 = 0.f, r2 = 0.f;
  const float* xrow;
  const float* x1row = x0;
  if (MODE == 0 || MODE == 1) {
    const int idxv = idx[qq * KN + lo];
    const float* pr = pcn + ((long long)b * NN + idxv) * 3;
    const float* pq = (MODE == 0) ? (pcn + ((long long)b * NN + qi) * 3)
                                  : (pcq + qq * 3);
    r0 = pr[0] - pq[0]; r1 = pr[1] - pq[1]; r2 = pr[2] - pq[2];
    xrow = x0 + ((long long)b * NN + idxv) * ((MODE == 0) ? Cx : CH);
    __builtin_prefetch(xrow, 0, 3);
  } else {
    xrow = x0 + (qq * KN + lo) * (long long)Cin;
    __builtin_prefetch(xrow, 0, 3);
  }
  if (MODE == 1) x1row = x1 + qq * CH;

  auto fetch = [&](int f) -> float {
    if (MODE == 0) {
      if (f < Cx) return xrow[f];
      const int r = f - Cx;
      return (r == 0) ? r0 : ((r == 1) ? r1 : ((r == 2) ? r2 : 0.f));
    } else if (MODE == 1) {
      if (f < CH)     return xrow[f];
      if (f < 2*CH)   return x1row[f - CH];
      const int r = f - 2*CH;
      return (r == 0) ? r0 : ((r == 1) ? r1 : ((r == 2) ? r2 : 0.f));
    } else {
      return xrow[f];   // dense modes: Cin is a multiple of 4
    }
  };

  constexpr int NT = COUT / 16;
  const v8f z8 = {0.f,0.f,0.f,0.f,0.f,0.f,0.f,0.f};
  v8f acc[NT];
#pragma unroll
  for (int t = 0; t < NT; ++t) acc[t] = z8;

  for (int k0 = 0; k0 < Cin4; k0 += 4) {
    const int f = k0 + 2 * hi;
    v2f a;
    a.x = fetch(f);
    a.y = fetch(f + 1);
#pragma unroll
    for (int t = 0; t < NT; ++t) {
      v2f bf;
      bf.x = Wl[f * COUT + t*16 + lo];
      bf.y = Wl[(f + 1) * COUT + t*16 + lo];
      acc[t] = wmma4(a, bf, acc[t]);
    }
  }

#pragma unroll
  for (int t = 0; t < NT; ++t) {
    const int col = t*16 + lo;
    const float bv = Bl[col];
    if (MODE == 0 || MODE == 3) {
      float m = -3.4e38f;
#pragma unroll
      for (int r = 0; r < 8; ++r) {
        float e = acc[t][r] + bv;
        e = e > 0.f ? e : 0.f;
        m = m > e ? m : e;
      }
      const float o = __shfl_xor(m, 16, 32);
      m = m > o ? m : o;
      if (hi == 0) out[qq * COUT + col] = m;
    } else {
#pragma unroll
      for (int r = 0; r < 8; ++r) {
        float e = acc[t][r] + bv;
        e = e > 0.f ? e : 0.f;
        out[(qq * KN + (r + 8*hi)) * (long long)COUT + col] = e;
      }
    }
  }
}

// ---------------------------------------------------------------------------
// sim[b] = f0[b] (4096x128) @ f1[b]^T.  128x64 tile / block, TDM-staged LDS.
// ---------------------------------------------------------------------------
__global__ __launch_bounds__(256) void gemm_nt_sim(
    const float* __restrict__ A, const float* __restrict__ Bm,
    float* __restrict__ out)
{
  extern __shared__ float lds[];
  float* As = lds;              // 128 x 128
  float* Bs = lds + 128 * 128;  // 64 x 128
  const int b  = blockIdx.z;
  const int m0 = blockIdx.y * 128;
  const int n0 = blockIdx.x * 64;
  const int tid = threadIdx.x;
  const float* Ab = A  + ((long long)b * NN + m0) * CH;
  const float* Bb = Bm + ((long long)b * NN + n0) * CH;
#if HAVE_TDM
  if (tid < 32) {
    tdm_load_1d(As, Ab, 128 * 128);
    tdm_load_1d(Bs, Bb, 64 * 128);
    tdm_wait();
  }
#else
  for (int t = tid; t < 128 * 128; t += 256) As[t] = Ab[t];
  for (int t = tid; t < 64 * 128; t += 256)  Bs[t] = Bb[t];
#endif
  __syncthreads();

  const int wave = tid >> 5, lane = tid & 31, lo = lane & 15, hi = lane >> 4;
  const v8f z8 = {0.f,0.f,0.f,0.f,0.f,0.f,0.f,0.f};
  v8f acc[4] = {z8, z8, z8, z8};
  const int ar = wave * 16 + lo;
  for (int k0 = 0; k0 < CH; k0 += 4) {
    const int kk = k0 + 2 * hi;
    v2f a; a.x = As[ar * CH + kk]; a.y = As[ar * CH + kk + 1];
#pragma unroll
    for (int t = 0; t < 4; ++t) {
      v2f bf;
      bf.x = Bs[(t*16 + lo) * CH + kk];
      bf.y = Bs[(t*16 + lo) * CH + kk + 1];
      acc[t] = wmma4(a, bf, acc[t]);
    }
  }
  const long long base = (long long)b * NN * NN;
#pragma unroll
  for (int t = 0; t < 4; ++t)
#pragma unroll
    for (int r = 0; r < 8; ++r)
      out[base + (long long)(m0 + wave*16 + r + 8*hi) * NN + (n0 + t*16 + lo)] = acc[t][r];
}

// ---------------------------------------------------------------------------
// Sinkhorn row/col logsumexp (online LSE + LDS tree reduce).
// ---------------------------------------------------------------------------
__global__ __launch_bounds__(256) void lse_row(
    const float* __restrict__ sim, const float* __restrict__ vv,
    float* __restrict__ uu, const float* __restrict__ alphaP,
    float norm_, float logTail)
{
  __shared__ float smx[256];
  __shared__ float ssm[256];
  const int i = blockIdx.x;           // 0..NN (row NN = dustbin row)
  const int b = blockIdx.y;
  const float alpha = alphaP[0];
  const float* vrow = vv + (long long)b * (NN + 1);
  const float* srow = sim + ((long long)b * NN + i) * NN;
  float mx = -3.4e38f, sm = 0.f;
  for (int j = threadIdx.x; j < NN; j += 256) {
    const float val = ((i < NN) ? srow[j] : alpha) + vrow[j];
    if (val > mx) { sm = sm * expf(mx - val) + 1.f; mx = val; }
    else          { sm += expf(val - mx); }
  }
  if (threadIdx.x == 0) {
    const float val = alpha + vrow[NN];
    if (val > mx) { sm = sm * expf(mx - val) + 1.f; mx = val; }
    else          { sm += expf(val - mx); }
  }
  smx[threadIdx.x] = mx; ssm[threadIdx.x] = sm;
  __syncthreads();
  for (int s = 128; s > 0; s >>= 1) {
    if (threadIdx.x < s) {
      const float m1 = smx[threadIdx.x], s1 = ssm[threadIdx.x];
      const float m2 = smx[threadIdx.x + s], s2 = ssm[threadIdx.x + s];
      const float m = m1 > m2 ? m1 : m2;
      ssm[threadIdx.x] = s1 * expf(m1 - m) + s2 * expf(m2 - m);
      smx[threadIdx.x] = m;
    }
    __syncthreads();
  }
  if (threadIdx.x == 0) {
    const float lse = smx[0] + logf(ssm[0]);
    const float lmu = (i < NN) ? norm_ : (logTail + norm_);
    uu[(long long)b * (NN + 1) + i] = lmu - lse;
  }
}

__global__ __launch_bounds__(256) void lse_col(
    const float* __restrict__ sim, const float* __restrict__ uu,
    float* __restrict__ vv, const float* __restrict__ alphaP,
    float norm_, float logTail)
{
  __shared__ float smx[256];
  __shared__ float ssm[256];
  const int j = blockIdx.x;
  const int b = blockIdx.y;
  const float alpha = alphaP[0];
  const float* urow = uu + (long long)b * (NN + 1);
  float mx = -3.4e38f, sm = 0.f;
  for (int i = threadIdx.x; i < NN; i += 256) {
    const float val = ((j < NN) ? sim[((long long)b * NN + i) * NN + j] : alpha) + urow[i];
    if (val > mx) { sm = sm * expf(mx - val) + 1.f; mx = val; }
    else          { sm += expf(val - mx); }
  }
  if (threadIdx.x == 0) {
    const float val = alpha + urow[NN];
    if (val > mx) { sm = sm * expf(mx - val) + 1.f; mx = val; }
    else          { sm += expf(val - mx); }
  }
  smx[threadIdx.x] = mx; ssm[threadIdx.x] = sm;
  __syncthreads();
  for (int s = 128; s > 0; s >>= 1) {
    if (threadIdx.x < s) {
      const float m1 = smx[threadIdx.x], s1 = ssm[threadIdx.x];
      const float m2 = smx[threadIdx.x + s], s2 = ssm[threadIdx.x + s];
      const float m = m1 > m2 ? m1 : m2;
      ssm[threadIdx.x] = s1 * expf(m1 - m) + s2 * expf(m2 - m);
      smx[threadIdx.x] = m;
    }
    __syncthreads();
  }
  if (threadIdx.x == 0) {
    const float lse = smx[0] + logf(ssm[0]);
    const float lnu = (j < NN) ? norm_ : (logTail + norm_);
    vv[(long long)b * (NN + 1) + j] = lnu - lse;
  }
}

// ---------------------------------------------------------------------------
// OT flow extraction: conf = exp(sim+u+v-norm)*[sqdist<1], fused reductions.
// ---------------------------------------------------------------------------
__global__ __launch_bounds__(256) void ot_row(
    const float* __restrict__ sim, const float* __restrict__ uu,
    const float* __restrict__ vv, const float* __restrict__ pc0,
    const float* __restrict__ pc1, float norm_, float* __restrict__ outF)
{
  __shared__ float sr[256], sx[256], sy[256], sz[256];
  const int i = blockIdx.x, b = blockIdx.y;
  const float ui = uu[(long long)b * (NN + 1) + i];
  const float* p0 = pc0 + ((long long)b * NN + i) * 3;
  const float qx = p0[0], qy = p0[1], qz = p0[2];
  const float* srow = sim + ((long long)b * NN + i) * NN;
  const float* p1b = pc1 + (long long)b * NN * 3;
  const float* vrow = vv + (long long)b * (NN + 1);
  float rs = 0.f, nx = 0.f, ny = 0.f, nz = 0.f;
  for (int j = threadIdx.x; j < NN; j += 256) {
    const float px = p1b[3*j], py = p1b[3*j+1], pz = p1b[3*j+2];
    const float dx = px - qx, dy = py - qy, dz = pz - qz;
    const float sq = dx*dx + dy*dy + dz*dz;
    const float a = expf(srow[j] + ui + vrow[j] - norm_);
    const float conf = (sq < 1.0f) ? a : 0.f;
    rs += conf; nx += conf * px; ny += conf * py; nz += conf * pz;
  }
  sr[threadIdx.x] = rs; sx[threadIdx.x] = nx; sy[threadIdx.x] = ny; sz[threadIdx.x] = nz;
  __syncthreads();
  for (int s = 128; s > 0; s >>= 1) {
    if (threadIdx.x < s) {
      sr[threadIdx.x] += sr[threadIdx.x + s];
      sx[threadIdx.x] += sx[threadIdx.x + s];
      sy[threadIdx.x] += sy[threadIdx.x + s];
      sz[threadIdx.x] += sz[threadIdx.x + s];
    }
    __syncthreads();
  }
  if (threadIdx.x == 0) {
    const float r = sr[0] + 1e-8f;
    const long long o = ((long long)b * NN + i) * 3;
    outF[o + 0] = sx[0] / r - qx;
    outF[o + 1] = sy[0] / r - qy;
    outF[o + 2] = sz[0] / r - qz;
  }
}

__global__ __launch_bounds__(256) void ot_col(
    const float* __restrict__ sim, const float* __restrict__ uu,
    const float* __restrict__ vv, const float* __restrict__ pc0,
    const float* __restrict__ pc1, float norm_, float* __restrict__ outF)
{
  __shared__ float sr[256], sx[256], sy[256], sz[256];
  const int j = blockIdx.x, b = blockIdx.y;
  const float vj = vv[(long long)b * (NN + 1) + j];
  const float* p1 = pc1 + ((long long)b * NN + j) * 3;
  const float qx = p1[0], qy = p1[1], qz = p1[2];
  const float* p0b = pc0 + (long long)b * NN * 3;
  const float* urow = uu + (long long)b * (NN + 1);
  float rs = 0.f, nx = 0.f, ny = 0.f, nz = 0.f;
  for (int i = threadIdx.x; i < NN; i += 256) {
    const float px = p0b[3*i], py = p0b[3*i+1], pz = p0b[3*i+2];
    const float dx = px - qx, dy = py - qy, dz = pz - qz;
    const float sq = dx*dx + dy*dy + dz*dz;
    const float a = expf(sim[((long long)b * NN + i) * NN + j] + urow[i] + vj - norm_);
    const float conf = (sq < 1.0f) ? a : 0.f;
    rs += conf; nx += conf * px; ny += conf * py; nz += conf * pz;
  }
  sr[threadIdx.x] = rs; sx[threadIdx.x] = nx; sy[threadIdx.x] = ny; sz[threadIdx.x] = nz;
  __syncthreads();
  for (int s = 128; s > 0; s >>= 1) {
    if (threadIdx.x < s) {
      sr[threadIdx.x] += sr[threadIdx.x + s];
      sx[threadIdx.x] += sx[threadIdx.x + s];
      sy[threadIdx.x] += sy[threadIdx.x + s];
      sz[threadIdx.x] += sz[threadIdx.x + s];
    }
    __syncthreads();
  }
  if (threadIdx.x == 0) {
    const float r = sr[0] + 1e-8f;
    const long long o = ((long long)b * NN + j) * 3;
    outF[o + 0] = sx[0] / r - qx;
    outF[o + 1] = sy[0] / r - qy;
    outF[o + 2] = sz[0] / r - qz;
  }
}

// ---------------------------------------------------------------------------
// Small dense per-point linear (fenc 3->128, fc 128->3, var_enc).
// act: 0 none, 1 relu, 2 sigmoid, 3 tanh
// ---------------------------------------------------------------------------
__global__ __launch_bounds__(256) void pointwise_linear(
    const float* __restrict__ x, const float* __restrict__ W,
    const float* __restrict__ bias, float* __restrict__ out,
    int Cin, int Cout, int act)
{
  const long long t = (long long)blockIdx.x * blockDim.x + threadIdx.x;
  const long long total = BN * (long long)Cout;
  if (t >= total) return;
  const int o = (int)(t % Cout);
  const long long p = t / Cout;
  const float* xr = x + p * Cin;
  float s = bias[o];
  for (int c = 0; c < Cin; ++c) s += xr[c] * W[c * Cout + o];
  if (act == 1)      s = s > 0.f ? s : 0.f;
  else if (act == 2) s = 1.f / (1.f + expf(-s));
  else if (act == 3) s = tanhf(s);
  out[p * Cout + o] = s;
}

// ---------------------------------------------------------------------------
// Elementwise helpers.
// ---------------------------------------------------------------------------
__global__ void ew_add(const float* a, const float* b, float* o, long long n) {
  long long t = (long long)blockIdx.x * blockDim.x + threadIdx.x;
  if (t < n) o[t] = a[t] + b[t];
}
__global__ void ew_sub(const float* a, const float* b, float* o, long long n) {
  long long t = (long long)blockIdx.x * blockDim.x + threadIdx.x;
  if (t < n) o[t] = a[t] - b[t];
}
__global__ void ew_mul(const float* a, const float* b, float* o, long long n) {
  long long t = (long long)blockIdx.x * blockDim.x + threadIdx.x;
  if (t < n) o[t] = a[t] * b[t];
}
__global__ void ew_sigmoid(const float* a, float* o, long long n) {
  long long t = (long long)blockIdx.x * blockDim.x + threadIdx.x;
  if (t < n) o[t] = 1.f / (1.f + expf(-a[t]));
}
__global__ void ew_tanh(const float* a, float* o, long long n) {
  long long t = (long long)blockIdx.x * blockDim.x + threadIdx.x;
  if (t < n) o[t] = tanhf(a[t]);
}
__global__ void ew_scale_rows(const float* x, const float* s, float* o, int Cc, long long n) {
  long long t = (long long)blockIdx.x * blockDim.x + threadIdx.x;
  if (t < n) o[t] = x[t] * s[t / Cc];
}
__global__ void ew_concat(const float* a, const float* b, float* o,
                          int C1, int C2, long long n) {
  long long t = (long long)blockIdx.x * blockDim.x + threadIdx.x;
  if (t >= n) return;
  const int Ct = C1 + C2;
  const long long p = t / Ct;
  const int c = (int)(t % Ct);
  o[t] = (c < C1) ? a[p * C1 + c] : b[p * C2 + (c - C1)];
}
__global__ void ew_gru(const float* z, const float* h, const float* q,
                       float* o, long long n) {
  long long t = (long long)blockIdx.x * blockDim.x + threadIdx.x;
  if (t < n) o[t] = (1.f - z[t]) * h[t] + z[t] * q[t];
}
__global__ void fill0(float* p, long long n) {
  long long t = (long long)blockIdx.x * blockDim.x + threadIdx.x;
  if (t < n) p[t] = 0.f;
}

// ---------------------------------------------------------------------------
// Host-side launch helpers.
// ---------------------------------------------------------------------------
static inline unsigned ewg(long long n) { return (unsigned)((n + 255) / 256); }

template<int COUT, int MODE>
static void launch_edge(const float* x0, const float* x1, const float* pcn,
                        const float* pcq, const int* idx, const float* W,
                        const float* bias, float* out, int Cx, int Cin,
                        hipStream_t stream)
{
  const int Cin4 = (Cin + 3) & ~3;
  const size_t lds = (size_t)(Cin4 * COUT + COUT) * sizeof(float);
  edge_mlp<COUT, MODE><<<dim3((unsigned)(BN / 8)), dim3(256), lds, stream>>>(
      x0, x1, pcn, pcq, idx, W, bias, out, Cx, Cin);
}

extern "C" void kernel_launch(void* const* d_in, const int* in_sizes, int n_in,
                              void* d_out, int out_size, void* d_ws, size_t ws_size,
                              hipStream_t stream)
{
  (void)in_sizes; (void)n_in; (void)out_size; (void)ws_size;
  const float* pc0 = (const float*)d_in[0];
  const float* pc1 = (const float*)d_in[1];
  const float* feat1_W = (const float*)d_in[2];  const float* feat1_b = (const float*)d_in[3];
  const float* feat2_W = (const float*)d_in[4];  const float* feat2_b = (const float*)d_in[5];
  const float* feat3_W = (const float*)d_in[6];  const float* feat3_b = (const float*)d_in[7];
  const float* flow1_W = (const float*)d_in[8];  const float* flow1_b = (const float*)d_in[9];
  const float* flow2_W = (const float*)d_in[10]; const float* flow2_b = (const float*)d_in[11];
  const float* fc_W    = (const float*)d_in[12]; const float* fc_b    = (const float*)d_in[13];
  const float* dx_W    = (const float*)d_in[14]; const float* dx_b    = (const float*)d_in[15];
  const float* fx_W    = (const float*)d_in[16]; const float* fx_b    = (const float*)d_in[17];
  const float* fenc_W  = (const float*)d_in[18]; const float* fenc_b  = (const float*)d_in[19];
  const float* h1_W    = (const float*)d_in[20]; const float* h1_b    = (const float*)d_in[21];
  const float* h2_W    = (const float*)d_in[22]; const float* h2_b    = (const float*)d_in[23];
  const float* gz_W    = (const float*)d_in[24]; const float* gz_b    = (const float*)d_in[25];
  const float* gr_W    = (const float*)d_in[26]; const float* gr_b    = (const float*)d_in[27];
  const float* gq_W    = (const float*)d_in[28]; const float* gq_b    = (const float*)d_in[29];
  const float* corr1_W = (const float*)d_in[30]; const float* corr1_b = (const float*)d_in[31];
  const float* corr2_W = (const float*)d_in[32]; const float* corr2_b = (const float*)d_in[33];
  const float* corr3_W = (const float*)d_in[34]; const float* corr3_b = (const float*)d_in[35];
  const float* var1_W  = (const float*)d_in[36]; const float* var1_b  = (const float*)d_in[37];
  const float* var2_W  = (const float*)d_in[38]; const float* var2_b  = (const float*)d_in[39];
  const float* bin_score = (const float*)d_in[40];

  float* outp = (float*)d_out;
  float* out0 = outp;                  // ot_flow0
  float* out1 = outp + BN * 3;         // flow0
  float* out2 = outp + 2 * BN * 3;     // ot_flow1
  float* out3 = outp + 3 * BN * 3;     // flow1

  // ---- workspace carve-up (floats, 256-aligned) ----
  float* ws = (float*)d_ws;
  size_t off = 0;
  auto alloc = [&](size_t n) { float* p = ws + off; off += (n + 255) & ~(size_t)255; return p; };
  const long long BNC = BN * CH;
  float* f0   = alloc(BNC);
  float* f1   = alloc(BNC);
  float* h0   = alloc(BNC);
  float* h1   = alloc(BNC);
  float* bufA = alloc(BN * 64);
  float* bufB = alloc(BN * 64);
  int*   idx0 = (int*)alloc(BN * KN);
  int*   idx1 = (int*)alloc(BN * KN);
  int*   idxw = (int*)alloc(BN * KN);
  float* uvec = alloc(BB * (NN + 1));
  float* vvec = alloc(BB * (NN + 1));
  float* warpb = alloc(BN * 3);
  float* Cm0  = alloc(BN);
  float* Cm1  = alloc(BN);
  float* tvar = alloc(BN * 32);
  float* pcw  = alloc(BN * 3);
  float* corrb = alloc(BNC);
  float* ffb  = alloc(BNC);
  float* xb   = alloc(BNC);
  float* hx   = alloc(BN * 2 * CH);
  float* zb   = alloc(BNC);
  float* rbuf = alloc(BNC);
  float* qb   = alloc(BNC);
  float* hn   = alloc(BNC);
  float* g1   = alloc(BNC);
  float* g2   = alloc(BNC);
  float* delta = alloc(BN * 3);
  float* sim  = alloc((size_t)BB * NN * NN);
  // after OT phase the sim region is reused for per-edge hiddens
  float* e1 = sim;
  float* e2 = sim + BN * KN * CH;

  const float norm_ = -logf((float)(2 * NN));  // -log(m+s)
  const float logT  = logf((float)NN);         // log(s) == log(m)

  // ---- graph KNN + feature pyramids ----
  knn_kernel<KN><<<ewg(BN), 256, 0, stream>>>(pc0, pc0, idx0);
  knn_kernel<KN><<<ewg(BN), 256, 0, stream>>>(pc1, pc1, idx1);

  launch_edge<32, 0>(pc0, nullptr, pc0, pc0, idx0, feat1_W, feat1_b, bufA, 3, 6, stream);
  launch_edge<64, 0>(bufA, nullptr, pc0, pc0, idx0, feat2_W, feat2_b, bufB, 32, 35, stream);
  launch_edge<128,0>(bufB, nullptr, pc0, pc0, idx0, feat3_W, feat3_b, f0, 64, 67, stream);
  launch_edge<32, 0>(pc1, nullptr, pc1, pc1, idx1, feat1_W, feat1_b, bufA, 3, 6, stream);
  launch_edge<64, 0>(bufA, nullptr, pc1, pc1, idx1, feat2_W, feat2_b, bufB, 32, 35, stream);
  launch_edge<128,0>(bufB, nullptr, pc1, pc1, idx1, feat3_W, feat3_b, f1, 64, 67, stream);

  // ---- similarity + Sinkhorn OT ----
  gemm_nt_sim<<<dim3(NN/64, NN/128, BB), dim3(256),
                (size_t)(128*128 + 64*128) * sizeof(float), stream>>>(f0, f1, sim);
  fill0<<<ewg(BB*(NN+1)), 256, 0, stream>>>(uvec, BB*(NN+1));
  fill0<<<ewg(BB*(NN+1)), 256, 0, stream>>>(vvec, BB*(NN+1));
  for (int it = 0; it < 2; ++it) {
    lse_row<<<dim3(NN+1, BB), 256, 0, stream>>>(sim, vvec, uvec, bin_score, norm_, logT);
    lse_col<<<dim3(NN+1, BB), 256, 0, stream>>>(sim, uvec, vvec, bin_score, norm_, logT);
  }
  ot_row<<<dim3(NN, BB), 256, 0, stream>>>(sim, uvec, vvec, pc0, pc1, norm_, out0);
  ot_col<<<dim3(NN, BB), 256, 0, stream>>>(sim, uvec, vvec, pc0, pc1, norm_, out2);

  // ---- hidden states ----
  launch_edge<128,0>(f0, nullptr, pc0, pc0, idx0, h1_W, h1_b, g1, CH, CH+3, stream);
  launch_edge<128,0>(g1, nullptr, pc0, pc0, idx0, h2_W, h2_b, h0, CH, CH+3, stream);
  ew_tanh<<<ewg(BNC), 256, 0, stream>>>(h0, h0, BNC);
  launch_edge<128,0>(f1, nullptr, pc1, pc1, idx1, h1_W, h1_b, g1, CH, CH+3, stream);
  launch_edge<128,0>(g1, nullptr, pc1, pc1, idx1, h2_W, h2_b, h1, CH, CH+3, stream);
  ew_tanh<<<ewg(BNC), 256, 0, stream>>>(h1, h1, BNC);

  // ---- confidence weights C0, C1 ----
  warp3_kernel<<<ewg(BN), 256, 0, stream>>>(pc0, pc1, out2, warpb);
  ew_sub<<<ewg(BN*3), 256, 0, stream>>>(out0, warpb, warpb, BN*3);
  pointwise_linear<<<ewg(BN*32), 256, 0, stream>>>(warpb, var1_W, var1_b, tvar, 3, 32, 0);
  pointwise_linear<<<ewg(BN), 256, 0, stream>>>(tvar, var2_W, var2_b, Cm0, 32, 1, 2);
  warp3_kernel<<<ewg(BN), 256, 0, stream>>>(pc1, pc0, out0, warpb);
  ew_sub<<<ewg(BN*3), 256, 0, stream>>>(out2, warpb, warpb, BN*3);
  pointwise_linear<<<ewg(BN*32), 256, 0, stream>>>(warpb, var1_W, var1_b, tvar, 3, 32, 0);
  pointwise_linear<<<ewg(BN), 256, 0, stream>>>(tvar, var2_W, var2_b, Cm1, 32, 1, 2);

  // ---- GRU refinement (one iteration), both directions ----
  auto refine = [&](const float* flow, float* flow_out, const float* Cm,
                    const float* pa, const float* pb, const float* fa,
                    const float* fb, const int* idxa, const float* hbuf)
  {
    ew_add<<<ewg(BN*3), 256, 0, stream>>>(pa, flow, pcw, BN*3);
    knn_kernel<KN><<<ewg(BN), 256, 0, stream>>>(pcw, pb, idxw);
    // FlowEmbedding: 3-layer edge MLP, max at the end
    launch_edge<128,1>(fb, fa, pb, pcw, idxw, corr1_W, corr1_b, e1, CH, 2*CH+3, stream);
    launch_edge<128,2>(e1, nullptr, nullptr, nullptr, nullptr, corr2_W, corr2_b, e2, CH, CH, stream);
    launch_edge<128,3>(e2, nullptr, nullptr, nullptr, nullptr, corr3_W, corr3_b, corrb, CH, CH, stream);
    ew_scale_rows<<<ewg(BNC), 256, 0, stream>>>(corrb, Cm, corrb, CH, BNC);
    launch_edge<128,0>(corrb, nullptr, pa, pa, idxa, dx_W, dx_b, xb, CH, CH+3, stream);
    pointwise_linear<<<ewg(BNC), 256, 0, stream>>>(flow, fenc_W, fenc_b, ffb, 3, CH, 0);
    ew_scale_rows<<<ewg(BNC), 256, 0, stream>>>(ffb, Cm, ffb, CH, BNC);
    launch_edge<128,0>(ffb, nullptr, pa, pa, idxa, fx_W, fx_b, g1, CH, CH+3, stream);
    ew_add<<<ewg(BNC), 256, 0, stream>>>(xb, g1, xb, BNC);
    // GRU
    ew_concat<<<ewg(BN*2*CH), 256, 0, stream>>>(hbuf, xb, hx, CH, CH, BN*2*CH);
    launch_edge<128,0>(hx, nullptr, pa, pa, idxa, gz_W, gz_b, zb, 2*CH, 2*CH+3, stream);
    ew_sigmoid<<<ewg(BNC), 256, 0, stream>>>(zb, zb, BNC);
    launch_edge<128,0>(hx, nullptr, pa, pa, idxa, gr_W, gr_b, rbuf, 2*CH, 2*CH+3, stream);
    ew_sigmoid<<<ewg(BNC), 256, 0, stream>>>(rbuf, rbuf, BNC);
    ew_mul<<<ewg(BNC), 256, 0, stream>>>(rbuf, hbuf, qb, BNC);
    ew_concat<<<ewg(BN*2*CH), 256, 0, stream>>>(qb, xb, hx, CH, CH, BN*2*CH);
    launch_edge<128,0>(hx, nullptr, pa, pa, idxa, gq_W, gq_b, qb, 2*CH, 2*CH+3, stream);
    ew_tanh<<<ewg(BNC), 256, 0, stream>>>(qb, qb, BNC);
    ew_gru<<<ewg(BNC), 256, 0, stream>>>(zb, hbuf, qb, hn, BNC);
    // flow head
    launch_edge<128,0>(hn, nullptr, pa, pa, idxa, flow1_W, flow1_b, g1, CH, CH+3, stream);
    launch_edge<128,0>(g1, nullptr, pa, pa, idxa, flow2_W, flow2_b, g2, CH, CH+3, stream);
    pointwise_linear<<<ewg(BN*3), 256, 0, stream>>>(g2, fc_W, fc_b, delta, CH, 3, 0);
    ew_add<<<ewg(BN*3), 256, 0, stream>>>(flow, delta, flow_out, BN*3);
  };

  refine(out0, out1, Cm0, pc0, pc1, f0, f1, idx0, h0);
  refine(out2, out3, Cm1, pc1, pc0, f1, f0, idx1, h1);
}